// CopyRNN_71313636983456
// MI455X (gfx1250) — compile-verified
//
#include <hip/hip_runtime.h>
#include <hip/hip_bf16.h>

// ---------------------------------------------------------------------------
// CopyRNN forward for MI455X (gfx1250, wave32, WMMA f32 16x16x4).
// ---------------------------------------------------------------------------
#define CB   16          // batch
#define CL   256         // seq len
#define CV   50000       // vocab
#define COOV 100
#define CE   128
#define CH   128
#define CVT  (CV + COOV) // 50100

typedef float v2f __attribute__((ext_vector_type(2)));
typedef float v8f __attribute__((ext_vector_type(8)));

__device__ __forceinline__ float sigm(float x) { return 1.0f / (1.0f + expf(-x)); }

// ---------------------------------------------------------------------------
// Generic WMMA GEMM:  C[M,N] = epi( A[M,K] @ W[N,K]^T  (+ bias[n]) )
// One wave per 16x16 C tile. epi: bit0 = +bias, bit1 = tanh.
// A/B fragment layout per CDNA5 ISA (16x4 A, 4x16 B, f32):
//   lane<16 holds K = {4k, 4k+1}, lane>=16 holds K = {4k+2, 4k+3}.
// ---------------------------------------------------------------------------
__global__ __launch_bounds__(128) void k_wmma_gemm(
    const float* __restrict__ A, const float* __restrict__ W,
    const float* __restrict__ bias, float* __restrict__ C,
    int M, int N, int K, int epi)
{
    const int lane = threadIdx.x & 31;
    const int wave = blockIdx.x * (blockDim.x >> 5) + (threadIdx.x >> 5);
    const int ntiles = (N + 15) >> 4;
    const int mtiles = (M + 15) >> 4;
    if (wave >= ntiles * mtiles) return;           // whole-wave exit: EXEC stays full for WMMA
    const int m0 = (wave / ntiles) * 16;
    const int n0 = (wave % ntiles) * 16;
    const int col = lane & 15;
    const int kb  = (lane < 16) ? 0 : 2;

    const int n  = n0 + col;
    const int nc = (n < N) ? n : (N - 1);          // clamped for safe loads
    const int m  = m0 + col;
    const int mc = (m < M) ? m : (M - 1);

    const float* __restrict__ arow = A + (size_t)mc * K;
    const float* __restrict__ wrow = W + (size_t)nc * K;

    v8f acc = {};
    for (int k = 0; k < K; k += 4) {
        v2f a = *(const v2f*)(arow + k + kb);
        v2f b = *(const v2f*)(wrow + k + kb);
        acc = __builtin_amdgcn_wmma_f32_16x16x4_f32(false, a, false, b,
                                                    (short)0, acc, false, false);
    }

    float bv = (epi & 1) ? bias[nc] : 0.0f;
    const int rbase = (lane < 16) ? 0 : 8;
#pragma unroll
    for (int r = 0; r < 8; ++r) {
        int row = m0 + r + rbase;
        if (row < M && n < N) {
            float v = acc[r] + bv;
            if (epi & 2) v = tanhf(v);
            C[(size_t)row * N + n] = v;
        }
    }
}

// ---------------------------------------------------------------------------
// Encoder LSTM recurrence: single workgroup, 32 waves, W_hh + h + gates in
// 296KB of CDNA5's 320KB LDS.  Wave w owns gate columns [16w,16w+16).
// Per step: g = xg[:,t,:] + h @ W_hh^T via 32 WMMAs/wave, then fused gates.
// ---------------------------------------------------------------------------
__global__ __launch_bounds__(1024) void k_enc_lstm(
    const float* __restrict__ xg,    // (B*L, 4H) = [(b*L+t)*512 + n]
    const float* __restrict__ W_hh,  // (512, 128)
    float* __restrict__ enc_out)     // (B*L, 128)
{
    extern __shared__ float smem[];
    float* sW = smem;                // 512*128 = 65536 f
    float* sH = sW + 512 * 128;      // 16*128  = 2048 f
    float* sG = sH + 16 * 128;       // 16*512  = 8192 f

    const int tid  = threadIdx.x;
    const int lane = tid & 31;
    const int wave = tid >> 5;       // 0..31 -> gate tile

    // Preload W_hh into LDS (float4 wide).
    for (int i = tid; i < (512 * 128) / 4; i += 1024)
        ((float4*)sW)[i] = ((const float4*)W_hh)[i];

    // Zero h; each thread owns two (b,j) cell states in registers.
    sH[tid] = 0.0f; sH[tid + 1024] = 0.0f;
    const int i0 = tid * 2, i1 = tid * 2 + 1;
    const int b0 = i0 >> 7, j0 = i0 & 127;
    const int b1 = i1 >> 7, j1 = i1 & 127;
    float c0 = 0.0f, c1 = 0.0f;
    __syncthreads();

    const int col   = lane & 15;
    const int kb    = (lane < 16) ? 0 : 2;
    const int n     = wave * 16 + col;
    const int rbase = (lane < 16) ? 0 : 8;
    const float* hrow = sH + col * 128;     // A: M = batch = col
    const float* wrow = sW + n * 128;       // B[k][n] = W_hh[n][k]

    for (int t = 0; t < CL; ++t) {
        v8f acc;
#pragma unroll
        for (int r = 0; r < 8; ++r)
            acc[r] = xg[((size_t)(r + rbase) * CL + t) * 512 + n];

#pragma unroll 4
        for (int k = 0; k < 128; k += 4) {
            v2f a = { hrow[k + kb], hrow[k + kb + 1] };
            v2f b = { wrow[k + kb], wrow[k + kb + 1] };
            acc = __builtin_amdgcn_wmma_f32_16x16x4_f32(false, a, false, b,
                                                        (short)0, acc, false, false);
        }
#pragma unroll
        for (int r = 0; r < 8; ++r)
            sG[(r + rbase) * 512 + n] = acc[r];
        __syncthreads();

        // Fused gate pointwise; h back to LDS + streamed to enc_out.
        {
            float gi = sG[b0 * 512 + j0], gf = sG[b0 * 512 + 128 + j0];
            float gg = sG[b0 * 512 + 256 + j0], go = sG[b0 * 512 + 384 + j0];
            c0 = sigm(gf) * c0 + sigm(gi) * tanhf(gg);
            float h = sigm(go) * tanhf(c0);
            sH[b0 * 128 + j0] = h;
            enc_out[((size_t)b0 * CL + t) * 128 + j0] = h;
        }
        {
            float gi = sG[b1 * 512 + j1], gf = sG[b1 * 512 + 128 + j1];
            float gg = sG[b1 * 512 + 256 + j1], go = sG[b1 * 512 + 384 + j1];
            c1 = sigm(gf) * c1 + sigm(gi) * tanhf(gg);
            float h = sigm(go) * tanhf(c1);
            sH[b1 * 128 + j1] = h;
            enc_out[((size_t)b1 * CL + t) * 128 + j1] = h;
        }
        __syncthreads();
    }
}

// ---------------------------------------------------------------------------
// Small helper kernels.
// ---------------------------------------------------------------------------
__global__ void k_embed(const int* __restrict__ tok, const float* __restrict__ emb,
                        float* __restrict__ x)
{
    int i = blockIdx.x * blockDim.x + threadIdx.x;      // B*L*E threads
    int h = i & 127, bl = i >> 7;
    x[i] = emb[(size_t)tok[bl] * CE + h];
}

__global__ void k_maskmul(const float* __restrict__ enc, const int* __restrict__ src,
                          const int* __restrict__ prev, float* __restrict__ mEnc)
{
    int i = blockIdx.x * blockDim.x + threadIdx.x;      // B*L*H threads
    int bl = i >> 7, b = bl >> 8;
    mEnc[i] = (prev[b] == src[bl]) ? enc[i] : 0.0f;
}

// Softmax over L with optional pad-mask, then ctx = sum_l w[l]*enc[b,l,:].
// logits[l] = rows[b,l,:] . vec[b,:]
__global__ __launch_bounds__(256) void k_softmax_ctx(
    const float* __restrict__ rows, const float* __restrict__ vec,
    const float* __restrict__ enc, const int* __restrict__ padtok,
    float* __restrict__ outv)
{
    __shared__ float wgt[CL];
    __shared__ float red[CL];
    const int b = blockIdx.x, l = threadIdx.x;
    const float NI = -__builtin_inff();

    const float* r = rows + ((size_t)b * CL + l) * CH;
    const float* v = vec + (size_t)b * CH;
    float s = 0.0f;
    for (int h = 0; h < CH; ++h) s += r[h] * v[h];
    if (padtok && padtok[b * CL + l] == 0) s = NI;

    red[l] = s; __syncthreads();
    for (int o = 128; o; o >>= 1) { if (l < o) red[l] = fmaxf(red[l], red[l + o]); __syncthreads(); }
    float mx = red[0]; __syncthreads();

    float e = expf(s - mx);
    wgt[l] = e; red[l] = e; __syncthreads();
    for (int o = 128; o; o >>= 1) { if (l < o) red[l] += red[l + o]; __syncthreads(); }
    float inv = 1.0f / red[0]; __syncthreads();
    wgt[l] = wgt[l] * inv; __syncthreads();

    if (l < CH) {
        float acc = 0.0f;
        for (int j = 0; j < CL; ++j)
            acc += wgt[j] * enc[((size_t)b * CL + j) * CH + l];
        outv[(size_t)b * CH + l] = acc;
    }
}

__global__ void k_dec_in(const int* __restrict__ prev, const float* __restrict__ emb,
                         const float* __restrict__ cst, float* __restrict__ dec_in)
{
    int i = blockIdx.x * blockDim.x + threadIdx.x;      // B*256 threads
    int b = i >> 8, j = i & 255;
    dec_in[i] = (j < CE) ? emb[(size_t)prev[b] * CE + j] : cst[b * CH + (j - CE)];
}

__global__ void k_dec_pointwise(const float* __restrict__ g, float* __restrict__ rnn_out)
{
    int i = blockIdx.x * blockDim.x + threadIdx.x;      // B*H = 2048 threads
    int b = i >> 7, j = i & 127;
    const float* gb = g + b * 512;
    float c = sigm(gb[j]) * tanhf(gb[256 + j]);          // h0 = c0 = 0
    rnn_out[i] = sigm(gb[384 + j]) * tanhf(c);
}

__global__ void k_cat2(const float* __restrict__ a, const float* __restrict__ b2,
                       float* __restrict__ c)
{
    int i = blockIdx.x * blockDim.x + threadIdx.x;      // B*256 threads
    int b = i >> 8, j = i & 255;
    c[i] = (j < CH) ? a[b * CH + j] : b2[b * CH + (j - CH)];
}

__global__ void k_zero(float* __restrict__ p, int n)
{
    int i = blockIdx.x * blockDim.x + threadIdx.x;
    if (i < n) p[i] = 0.0f;
}

// Sparse copy-score: agg is zero except at tokens present in src row, and
// tanh(0)=0 -> score 0 elsewhere. One block per (b,l); duplicates compute
// bit-identical values so plain stores are race-free.
__global__ __launch_bounds__(128) void k_copy_sparse(
    const int* __restrict__ toks, const float* __restrict__ enc,
    const float* __restrict__ copy_W, const float* __restrict__ attn_out,
    float* __restrict__ cs)
{
    __shared__ int   st[CL];
    __shared__ float aggr[CH];
    __shared__ float red[CH];
    const int bl = blockIdx.x, b = bl >> 8, l = bl & 255;
    const int tid = threadIdx.x;

    st[tid]       = toks[b * CL + tid];
    st[tid + 128] = toks[b * CL + tid + 128];
    __syncthreads();
    const int tok = st[l];

    float s = 0.0f;                                      // agg row, h = tid
    for (int l2 = 0; l2 < CL; ++l2)
        if (st[l2] == tok) s += enc[((size_t)b * CL + l2) * CH + tid];
    aggr[tid] = s; __syncthreads();

    float t = 0.0f;                                      // g = tid
    for (int h = 0; h < CH; ++h) t += aggr[h] * copy_W[tid * CH + h];
    red[tid] = attn_out[b * CH + tid] * tanhf(t); __syncthreads();
    for (int o = 64; o; o >>= 1) { if (tid < o) red[tid] += red[tid + o]; __syncthreads(); }
    if (tid == 0) cs[(size_t)b * CVT + tok] = red[0];
}

// Final masked softmax over V+OOV per batch row.
__global__ __launch_bounds__(1024) void k_total_softmax(
    const float* __restrict__ gen, const float* __restrict__ cs,
    const int* __restrict__ oovc, float* __restrict__ prob)
{
    __shared__ float red[1024];
    const int b = blockIdx.x, tid = threadIdx.x;
    const int lim = CV + oovc[b];
    const float NI = -__builtin_inff();
    const float* gb = gen + (size_t)b * CVT;
    const float* cb = cs  + (size_t)b * CVT;

    float mx = NI;
    for (int v = tid; v < CVT; v += 1024) {
        float val = gb[v] + cb[v];
        if (v >= lim || v == 1) val = NI;
        mx = fmaxf(mx, val);
    }
    red[tid] = mx; __syncthreads();
    for (int o = 512; o; o >>= 1) { if (tid < o) red[tid] = fmaxf(red[tid], red[tid + o]); __syncthreads(); }
    mx = red[0]; __syncthreads();

    float s = 0.0f;
    for (int v = tid; v < CVT; v += 1024) {
        float val = gb[v] + cb[v];
        if (v >= lim || v == 1) val = NI;
        s += expf(val - mx);
    }
    red[tid] = s; __syncthreads();
    for (int o = 512; o; o >>= 1) { if (tid < o) red[tid] += red[tid + o]; __syncthreads(); }
    const float inv = 1.0f / red[0];

    float* pb = prob + (size_t)b * CVT;
    for (int v = tid; v < CVT; v += 1024) {
        float val = gb[v] + cb[v];
        if (v >= lim || v == 1) val = NI;
        pb[v] = expf(val - mx) * inv;
    }
}

// ---------------------------------------------------------------------------
static inline int gemmBlocks(int M, int N) {
    int tiles = ((M + 15) / 16) * ((N + 15) / 16);
    return (tiles + 3) / 4;                              // 4 waves / 128-thread block
}

extern "C" void kernel_launch(void* const* d_in, const int* in_sizes, int n_in,
                              void* d_out, int out_size, void* d_ws, size_t ws_size,
                              hipStream_t stream)
{
    (void)in_sizes; (void)n_in; (void)out_size; (void)ws_size;
    const int*   src_tokens = (const int*)  d_in[0];
    const int*   prev_out   = (const int*)  d_in[2];
    const int*   src_oov    = (const int*)  d_in[4];
    const int*   oovc       = (const int*)  d_in[5];
    const float* dec_state  = (const float*)d_in[6];
    const float* emb        = (const float*)d_in[7];
    const float* enc_W_ih   = (const float*)d_in[8];
    const float* enc_W_hh   = (const float*)d_in[9];
    const float* enc_b      = (const float*)d_in[10];
    const float* dec_W_ih   = (const float*)d_in[11];
    const float* dec_b      = (const float*)d_in[13];
    const float* attn_W     = (const float*)d_in[14];
    const float* out_W      = (const float*)d_in[15];
    const float* out_b      = (const float*)d_in[16];
    const float* copy_W     = (const float*)d_in[17];
    const float* gen_W      = (const float*)d_in[18];

    float* out        = (float*)d_out;
    float* total_prob = out;                                   // B*VT
    float* enc_out    = out + (size_t)CB * CVT;                // B*L*H
    float* attn_out   = enc_out + (size_t)CB * CL * CH;        // B*H

    float* w = (float*)d_ws;
    float* x     = w; w += (size_t)CB * CL * CE;
    float* xg    = w; w += (size_t)CB * CL * 4 * CH;
    float* mEnc  = w; w += (size_t)CB * CL * CH;
    float* T1    = w; w += (size_t)CB * CL * CH;
    float* cst   = w; w += CB * CH;
    float* decin = w; w += CB * (CE + CH);
    float* g_dec = w; w += CB * 4 * CH;
    float* rnn   = w; w += CB * CH;
    float* q     = w; w += CB * CH;
    float* ctx   = w; w += CB * CH;
    float* cat2  = w; w += CB * 2 * CH;
    float* genb  = w; w += (size_t)CB * CVT;
    float* cs    = w; w += (size_t)CB * CVT;

    const int BL = CB * CL;                                    // 4096

    // 1) x = embedding[src_tokens]
    k_embed<<<(BL * CE) / 256, 256, 0, stream>>>(src_tokens, emb, x);
    // 2) xg = x @ enc_W_ih^T + enc_b     (4096 x 512 x 128, WMMA)
    k_wmma_gemm<<<gemmBlocks(BL, 512), 128, 0, stream>>>(x, enc_W_ih, enc_b, xg,
                                                         BL, 512, CE, 1);
    // 3) encoder recurrence (single WGP, W_hh in LDS, WMMA per step)
    const size_t smem = (512 * 128 + 16 * 128 + 16 * 512) * sizeof(float); // ~296KB
    k_enc_lstm<<<1, 1024, smem, stream>>>(xg, enc_W_hh, enc_out);
    // 4) masked encoder states
    k_maskmul<<<(BL * CH) / 256, 256, 0, stream>>>(enc_out, src_tokens, prev_out, mEnc);
    // 5) agg_w = tanh(mEnc @ copy_W^T)
    k_wmma_gemm<<<gemmBlocks(BL, CH), 128, 0, stream>>>(mEnc, copy_W, nullptr, T1,
                                                        BL, CH, CH, 2);
    // 6) copy attention softmax + copy_state
    k_softmax_ctx<<<CB, CL, 0, stream>>>(T1, dec_state, enc_out, nullptr, cst);
    // 7) decoder input = [embed(prev), copy_state]
    k_dec_in<<<(CB * 256) / 256, 256, 0, stream>>>(prev_out, emb, cst, decin);
    // 8) decoder gates = dec_in @ dec_W_ih^T + dec_b (h0 = 0 -> no W_hh term)
    k_wmma_gemm<<<gemmBlocks(CB, 512), 128, 0, stream>>>(decin, dec_W_ih, dec_b, g_dec,
                                                         CB, 512, CE + CH, 1);
    k_dec_pointwise<<<(CB * CH) / 256, 256, 0, stream>>>(g_dec, rnn);
    // 9) q = rnn_out @ attn_W^T
    k_wmma_gemm<<<gemmBlocks(CB, CH), 128, 0, stream>>>(rnn, attn_W, nullptr, q,
                                                        CB, CH, CH, 0);
    // 10) attention softmax (pad-masked) + ctx
    k_softmax_ctx<<<CB, CL, 0, stream>>>(enc_out, q, enc_out, src_tokens, ctx);
    // 11) attn_out = tanh([rnn, ctx] @ out_W^T + out_b)  -> directly into d_out
    k_cat2<<<(CB * 256) / 256, 256, 0, stream>>>(rnn, ctx, cat2);
    k_wmma_gemm<<<gemmBlocks(CB, CH), 128, 0, stream>>>(cat2, out_W, out_b, attn_out,
                                                        CB, CH, 2 * CH, 3);
    // 12) gen_logits = attn_out @ gen_W^T   (16 x 50100 x 128, bandwidth-bound)
    k_wmma_gemm<<<gemmBlocks(CB, CVT), 128, 0, stream>>>(attn_out, gen_W, nullptr, genb,
                                                         CB, CVT, CH, 0);
    // 13) sparse copy_score (zero everywhere except source tokens)
    k_zero<<<(CB * CVT + 1023) / 1024, 1024, 0, stream>>>(cs, CB * CVT);
    k_copy_sparse<<<BL, 128, 0, stream>>>(src_oov, enc_out, copy_W, attn_out, cs);
    // 14) final masked softmax over V+OOV
    k_total_softmax<<<CB, 1024, 0, stream>>>(genb, cs, oovc, total_prob);
}